// FSCLoss3D_48095043781220
// MI455X (gfx1250) — compile-verified
//
#include <hip/hip_runtime.h>
#include <math.h>

// ---------------------------------------------------------------------------
// FSC loss: 3D FFT via DFT-as-GEMM on fp32 WMMA (v_wmma_f32_16x16x4_f32),
// LDS-staged B tiles shared by all 8 waves of a block, 2 N-tiles per wave,
// then per-batch shell-binned power sums and the scalar loss.
// ---------------------------------------------------------------------------

typedef __attribute__((ext_vector_type(2))) float v2f;
typedef __attribute__((ext_vector_type(8))) float v8f;

#define FSC_SIZE   128
#define FSC_V      (FSC_SIZE * FSC_SIZE * FSC_SIZE)   // 2,097,152
#define FSC_NCOLS  (FSC_SIZE * FSC_SIZE)              // 16,384 columns per pass
#define FSC_NSEG   65                                  // shells 0..63 used, 64 = overflow
#define FSC_NB     8
#define PADK       132   // padded column stride in LDS: 132 mod 64 = 4 -> banks 4c+k,
                         // conflict-free for 16 cols x b64; 528B stride is 16B aligned

static __device__ __forceinline__ v8f wmma_f32(v2f a, v2f b, v8f c) {
  // D(16x16,f32) = A(16x4,f32) x B(4x16,f32) + C   -- fp32 matrix pipe
  return __builtin_amdgcn_wmma_f32_16x16x4_f32(false, a, false, b,
                                               (short)0, c, false, false);
}

// Build DFT matrix W[f][k] = exp(-2*pi*i*f*k/128). Modular reduction of f*k
// keeps the angle in [-2pi, 0) for full fp32 accuracy.
__global__ void fsc_build_dft(float* __restrict__ Wr, float* __restrict__ Wi) {
  int i = blockIdx.x * blockDim.x + threadIdx.x;     // 16384 threads
  int f = i >> 7;
  int k = i & 127;
  int fk = (f * k) & 127;
  float ang = -2.0f * 3.14159265358979323846f * (float)fk * (1.0f / 128.0f);
  Wr[i] = cosf(ang);
  Wi[i] = sinf(ang);
}

__global__ void fsc_zero(float* __restrict__ p, int n) {
  int i = blockIdx.x * blockDim.x + threadIdx.x;
  if (i < n) p[i] = 0.0f;
}

// ---------------------------------------------------------------------------
// Pass kernels. Data matrix: column c (0..16383) holds 128 contiguous floats,
// element (k, c) at in[c*128 + k]. Output row-major: out[f*16384 + c] -- this
// is exactly the axis rotation that makes the next axis contiguous.
//
// Block = 256 threads = 8 waves. Block covers M-tiles 0..7 (wave id) and two
// N-tiles (32 columns). B (32 cols x 128 K) is staged once into LDS with
// padded stride PADK; A rows come from the L2-resident W matrices.
//
// WMMA f32 16x16x4 lane layout (wave32):
//   A 16x4 : lane<16 -> row=lane,   K={k0,k0+1};  lane>=16 -> row=lane-16, K={k0+2,k0+3}
//   B 4x16 : lane<16 -> col=lane,   K={k0,k0+1};  lane>=16 -> col=lane-16, K={k0+2,k0+3}
//   C/D    : n = lane&15, m = vgpr + 8*(lane>=16)
// ---------------------------------------------------------------------------

// Pass 1: real input -> complex output. 4 WMMA per k-step (2 tiles x 2).
__global__ __launch_bounds__(256) void fsc_pass_r2c(
    const float* __restrict__ x,
    float* __restrict__ outR, float* __restrict__ outI,
    const float* __restrict__ Wr, const float* __restrict__ Wi) {
  __shared__ float sX[32 * PADK];
  const int t = threadIdx.x;
  const int colbase = blockIdx.x * 32;

  // Stage 32 columns (32*128 floats = 1024 float4) cooperatively.
  {
    const float4* g = (const float4*)(x + (size_t)colbase * FSC_SIZE);
#pragma unroll
    for (int r = 0; r < 4; ++r) {
      int q   = r * 256 + t;          // float4 index
      int col = q >> 5;               // 32 float4 per column
      int off = (q & 31) << 2;
      *(float4*)&sX[col * PADK + off] = g[q];
    }
  }
  __syncthreads();

  const int lane = t & 31;
  const int mt   = t >> 5;                    // wave id = M tile
  const int l15  = lane & 15;
  const int hi   = lane >> 4;
  const int koff = hi << 1;
  const int arow = (mt * 16 + l15) * FSC_SIZE;
  const float* b0 = &sX[l15 * PADK];          // N-tile 0, this lane's column
  const float* b1 = &sX[(16 + l15) * PADK];   // N-tile 1

  v8f aR0 = {}, aI0 = {}, aR1 = {}, aI1 = {};
#pragma unroll 2
  for (int k0 = 0; k0 < FSC_SIZE; k0 += 4) {
    const int ka = k0 + koff;
    v2f ar  = *(const v2f*)(Wr + arow + ka);
    v2f ai  = *(const v2f*)(Wi + arow + ka);
    v2f bx0 = *(const v2f*)(b0 + ka);
    v2f bx1 = *(const v2f*)(b1 + ka);
    aR0 = wmma_f32(ar, bx0, aR0);
    aI0 = wmma_f32(ai, bx0, aI0);
    aR1 = wmma_f32(ar, bx1, aR1);
    aI1 = wmma_f32(ai, bx1, aI1);
  }
  const int mbase = mt * 16 + (hi << 3);
  const int n0    = colbase + l15;
#pragma unroll
  for (int i = 0; i < 8; ++i) {
    const int row = (mbase + i) * FSC_NCOLS;
    outR[row + n0]      = aR0[i];
    outI[row + n0]      = aI0[i];
    outR[row + n0 + 16] = aR1[i];
    outI[row + n0 + 16] = aI1[i];
  }
}

// Passes 2/3: complex -> complex. 8 WMMA per k-step (2 tiles x 4).
// Re = Wr*xr - Wi*xi (two accumulators; f32 WMMA has no A/B negate)
// Im = Wr*xi + Wi*xr (one shared accumulator)
__global__ __launch_bounds__(256) void fsc_pass_c2c(
    const float* __restrict__ inR, const float* __restrict__ inI,
    float* __restrict__ outR, float* __restrict__ outI,
    const float* __restrict__ Wr, const float* __restrict__ Wi) {
  __shared__ float sB[2 * 32 * PADK];
  float* sR = sB;
  float* sI = sB + 32 * PADK;
  const int t = threadIdx.x;
  const int colbase = blockIdx.x * 32;

  {
    const float4* gR = (const float4*)(inR + (size_t)colbase * FSC_SIZE);
    const float4* gI = (const float4*)(inI + (size_t)colbase * FSC_SIZE);
#pragma unroll
    for (int r = 0; r < 4; ++r) {
      int q   = r * 256 + t;
      int col = q >> 5;
      int off = (q & 31) << 2;
      *(float4*)&sR[col * PADK + off] = gR[q];
      *(float4*)&sI[col * PADK + off] = gI[q];
    }
  }
  __syncthreads();

  const int lane = t & 31;
  const int mt   = t >> 5;
  const int l15  = lane & 15;
  const int hi   = lane >> 4;
  const int koff = hi << 1;
  const int arow = (mt * 16 + l15) * FSC_SIZE;
  const float* bR0 = &sR[l15 * PADK];
  const float* bR1 = &sR[(16 + l15) * PADK];
  const float* bI0 = &sI[l15 * PADK];
  const float* bI1 = &sI[(16 + l15) * PADK];

  v8f r10 = {}, r20 = {}, im0 = {};   // N-tile 0
  v8f r11 = {}, r21 = {}, im1 = {};   // N-tile 1
#pragma unroll 2
  for (int k0 = 0; k0 < FSC_SIZE; k0 += 4) {
    const int ka = k0 + koff;
    v2f ar  = *(const v2f*)(Wr + arow + ka);
    v2f ai  = *(const v2f*)(Wi + arow + ka);
    v2f br0 = *(const v2f*)(bR0 + ka);
    v2f bi0 = *(const v2f*)(bI0 + ka);
    v2f br1 = *(const v2f*)(bR1 + ka);
    v2f bi1 = *(const v2f*)(bI1 + ka);
    r10 = wmma_f32(ar, br0, r10);
    r20 = wmma_f32(ai, bi0, r20);
    im0 = wmma_f32(ar, bi0, im0);
    im0 = wmma_f32(ai, br0, im0);
    r11 = wmma_f32(ar, br1, r11);
    r21 = wmma_f32(ai, bi1, r21);
    im1 = wmma_f32(ar, bi1, im1);
    im1 = wmma_f32(ai, br1, im1);
  }
  const int mbase = mt * 16 + (hi << 3);
  const int n0    = colbase + l15;
#pragma unroll
  for (int i = 0; i < 8; ++i) {
    const int row = (mbase + i) * FSC_NCOLS;
    outR[row + n0]      = r10[i] - r20[i];
    outI[row + n0]      = im0[i];
    outR[row + n0 + 16] = r11[i] - r21[i];
    outI[row + n0 + 16] = im1[i];
  }
}

// Per-batch shell accumulation: LDS-staged segment sums, one global atomic
// per segment per block. acc layout per batch: [cross(65) | p1(65) | p2(65)].
__global__ __launch_bounds__(256) void fsc_shell_acc(
    const float* __restrict__ f1r, const float* __restrict__ f1i,
    const float* __restrict__ f2r, const float* __restrict__ f2i,
    const int* __restrict__ sidx, float* __restrict__ acc) {
  __shared__ float s[3 * FSC_NSEG];
  if (threadIdx.x < 3 * FSC_NSEG) s[threadIdx.x] = 0.0f;
  __syncthreads();

  const int v = blockIdx.x * blockDim.x + threadIdx.x;   // grid covers FSC_V exactly
  const float a = f1r[v], b = f1i[v], c = f2r[v], d = f2i[v];
  const float cross = a * c + b * d;          // Re(F1 * conj(F2))
  const float p1    = a * a + b * b;
  const float p2    = c * c + d * d;
  const int   seg   = sidx[v];                // axis-symmetric -> layout-safe
  atomicAdd(&s[seg], cross);
  atomicAdd(&s[FSC_NSEG + seg], p1);
  atomicAdd(&s[2 * FSC_NSEG + seg], p2);
  __syncthreads();

  if (threadIdx.x < 3 * FSC_NSEG) atomicAdd(&acc[threadIdx.x], s[threadIdx.x]);
}

// Final: fsc = cross / (sqrt(p1*p2) + eps) per (batch, shell<64);
// loss = 1 - mean(fsc^2) over 8*64 values.
__global__ __launch_bounds__(512) void fsc_finalize(const float* __restrict__ acc,
                                                    float* __restrict__ out) {
  __shared__ float red[512];
  const int t  = threadIdx.x;          // 512 = 8 batches * 64 shells
  const int b  = t >> 6;
  const int sh = t & 63;
  const float* a = acc + b * 3 * FSC_NSEG;
  const float num = a[sh];
  const float d1  = a[FSC_NSEG + sh];
  const float d2  = a[2 * FSC_NSEG + sh];
  const float fsc = num / (sqrtf(d1 * d2) + 1e-8f);
  red[t] = fsc * fsc;
  __syncthreads();
#pragma unroll
  for (int stp = 256; stp > 0; stp >>= 1) {
    if (t < stp) red[t] += red[t + stp];
    __syncthreads();
  }
  if (t == 0) out[0] = 1.0f - red[0] * (1.0f / 512.0f);
}

// ---------------------------------------------------------------------------
// Host-side launcher. Workspace layout (floats):
//   [0..16383]        Wr
//   [16384..32767]    Wi
//   [32768..34327]    acc (8 batches * 195), padded to 36864
//   [36864..]         6 volume buffers of FSC_V floats each (A/B/C re+im)
// Total ~50.5 MB.
// ---------------------------------------------------------------------------
extern "C" void kernel_launch(void* const* d_in, const int* in_sizes, int n_in,
                              void* d_out, int out_size, void* d_ws, size_t ws_size,
                              hipStream_t stream) {
  const float* img_ref  = (const float*)d_in[0];
  const float* img_pred = (const float*)d_in[1];
  const int*   sidx     = (const int*)d_in[2];
  float*       out      = (float*)d_out;
  float*       ws       = (float*)d_ws;

  float* Wr  = ws;
  float* Wi  = ws + 16384;
  float* acc = ws + 32768;
  float* buf = ws + 36864;
  float* Are = buf + (size_t)0 * FSC_V;
  float* Aim = buf + (size_t)1 * FSC_V;
  float* Bre = buf + (size_t)2 * FSC_V;
  float* Bim = buf + (size_t)3 * FSC_V;
  float* Cre = buf + (size_t)4 * FSC_V;
  float* Cim = buf + (size_t)5 * FSC_V;

  fsc_build_dft<<<64, 256, 0, stream>>>(Wr, Wi);
  fsc_zero<<<(FSC_NB * 3 * FSC_NSEG + 255) / 256, 256, 0, stream>>>(
      acc, FSC_NB * 3 * FSC_NSEG);

  const int passGrid  = FSC_NCOLS / 32;   // 512 blocks: 8 M-tiles x 2 N-tiles each
  const int shellGrid = FSC_V / 256;      // 8192

  for (int b = 0; b < FSC_NB; ++b) {
    const float* r = img_ref  + (size_t)b * FSC_V;
    const float* p = img_pred + (size_t)b * FSC_V;
    // ref: in -> A -> B -> A   (final ref spectrum in A)
    fsc_pass_r2c<<<passGrid, 256, 0, stream>>>(r, Are, Aim, Wr, Wi);
    fsc_pass_c2c<<<passGrid, 256, 0, stream>>>(Are, Aim, Bre, Bim, Wr, Wi);
    fsc_pass_c2c<<<passGrid, 256, 0, stream>>>(Bre, Bim, Are, Aim, Wr, Wi);
    // pred: in -> B -> C -> B  (final pred spectrum in B)
    fsc_pass_r2c<<<passGrid, 256, 0, stream>>>(p, Bre, Bim, Wr, Wi);
    fsc_pass_c2c<<<passGrid, 256, 0, stream>>>(Bre, Bim, Cre, Cim, Wr, Wi);
    fsc_pass_c2c<<<passGrid, 256, 0, stream>>>(Cre, Cim, Bre, Bim, Wr, Wi);
    // shell sums for this batch
    fsc_shell_acc<<<shellGrid, 256, 0, stream>>>(Are, Aim, Bre, Bim, sidx,
                                                 acc + b * 3 * FSC_NSEG);
  }
  fsc_finalize<<<1, 512, 0, stream>>>(acc, out);
}